// GNN_18829136626166
// MI455X (gfx1250) — compile-verified
//
#include <hip/hip_runtime.h>

typedef __attribute__((ext_vector_type(2))) float v2f;
typedef __attribute__((ext_vector_type(8))) float v8f;

#define FEAT 128

// ---------------- degree / normalization ----------------

__global__ __launch_bounds__(256) void k_fill_deg(float* __restrict__ deg, int n) {
    int i = blockIdx.x * 256 + threadIdx.x;
    if (i < n) deg[i] = 1.0f;               // self-loop contribution
}

__global__ __launch_bounds__(256) void k_deg_edges(float* __restrict__ deg,
                                                   const int* __restrict__ dst, int e) {
    int i = blockIdx.x * 256 + threadIdx.x;
    if (i < e)
        __hip_atomic_fetch_add(&deg[dst[i]], 1.0f, __ATOMIC_RELAXED, __HIP_MEMORY_SCOPE_AGENT);
}

__global__ __launch_bounds__(256) void k_deg_to_dinv(float* __restrict__ d, int n) {
    int i = blockIdx.x * 256 + threadIdx.x;
    if (i < n) {
        float v = d[i];
        d[i] = (v > 0.0f) ? rsqrtf(v) : 0.0f;
    }
}

// ---------------- fp32 WMMA GEMM: D[M,128] = A[M,128] * B(128,128) ----------------
// B is staged once per block into 64KB LDS as (k,k+1) float2 pairs with an XOR
// swizzle so the 32-lane ds_load_b64 in the inner loop is bank-conflict-free.
// Block = 8 waves = one 16-row M strip x all 8 N tiles; persistent over mTiles.
// TRANSB=0: B row-major [K][N] (GCN weight w). TRANSB=1: B is lw[N][K] (A*lw^T).

template<bool TRANSB, bool RELU, bool BIAS>
__global__ __launch_bounds__(256) void k_gemm_wmma(const float* __restrict__ A,
                                                   const float* __restrict__ B,
                                                   const float* __restrict__ bias,
                                                   float* __restrict__ D, int M) {
    __shared__ float sB[64 * FEAT * 2];   // 65536 B: [k/2][col] float2 pairs, swizzled

    // ---- stage B into LDS (coalesced global reads for both layouts) ----
    for (int idx = threadIdx.x; idx < FEAT * FEAT; idx += 256) {
        int k, col;
        float v;
        if (TRANSB) {
            col = idx >> 7; k = idx & 127;          // read lw[col][k] rows coalesced
            v = B[(size_t)col * FEAT + k];
        } else {
            k = idx >> 7; col = idx & 127;          // read w[k][col] coalesced
            v = B[idx];
        }
        int p = k >> 1;                             // pair index 0..63
        int e = p * FEAT + (col ^ ((p & 1) << 4));  // XOR swizzle on odd pairs
        sB[e * 2 + (k & 1)] = v;
    }
    __syncthreads();

    int wave = threadIdx.x >> 5;   // nTile
    int lane = threadIdx.x & 31;
    int half = lane >> 4;          // 0: K=+0,+1 ; 1: K=+2,+3 within each 4-wide K step
    int r    = lane & 15;
    int col  = wave * 16 + r;
    int mTiles = (M + 15) >> 4;

    for (int m = blockIdx.x; m < mTiles; m += gridDim.x) {
        int row = m * 16 + r;
        if (row >= M) row = M - 1;                  // keep EXEC full for WMMA
        const float* arow = A + (size_t)row * FEAT;

        v8f c = {};
#pragma unroll
        for (int k = 0; k < FEAT; k += 4) {
            int kk = k + 2 * half;
            v2f a = *(const v2f*)(arow + kk);
            int p = kk >> 1;                        // (p & 1) == half here
            v2f b = *(const v2f*)(sB + (size_t)(p * FEAT + (col ^ (half << 4))) * 2);
            c = __builtin_amdgcn_wmma_f32_16x16x4_f32(false, a, false, b,
                                                      (short)0, c, false, false);
        }

        float bv = BIAS ? bias[col] : 0.0f;
#pragma unroll
        for (int i = 0; i < 8; ++i) {
            int orow = m * 16 + half * 8 + i;       // D layout: vgpr i -> M = i + 8*half
            float v = c[i] + bv;
            if (RELU) v = fmaxf(v, 0.0f);
            if (orow < M) D[(size_t)orow * FEAT + col] = v;
        }
    }
}

// ---------------- agg init: agg = cb + h * dinv^2 (self-loop folded in) ----------------

__global__ __launch_bounds__(256) void k_init_agg(float* __restrict__ agg,
                                                  const float* __restrict__ h,
                                                  const float* __restrict__ dinv,
                                                  const float* __restrict__ cb, int n) {
    long i = (long)blockIdx.x * 256 + threadIdx.x;
    if (i < (long)n * FEAT) {
        float d = dinv[i >> 7];
        agg[i] = cb[i & (FEAT - 1)] + h[i] * d * d;
    }
}

// ---------------- edge scatter: agg[dst] += h[src] * dinv[src]*dinv[dst] ----------------
// one wave per edge, float4 per lane (32*4 = 128 floats)

__global__ __launch_bounds__(256) void k_scatter(float* __restrict__ agg,
                                                 const float* __restrict__ h,
                                                 const float* __restrict__ dinv,
                                                 const int* __restrict__ src,
                                                 const int* __restrict__ dst, int e) {
    int wid  = (blockIdx.x * 256 + threadIdx.x) >> 5;
    int lane = threadIdx.x & 31;
    if (wid >= e) return;
    int s = src[wid];
    int d = dst[wid];
    float w = dinv[s] * dinv[d];
    float4 v = *((const float4*)(h + (size_t)s * FEAT) + lane);
    float* ap = agg + (size_t)d * FEAT + lane * 4;
    __hip_atomic_fetch_add(ap + 0, v.x * w, __ATOMIC_RELAXED, __HIP_MEMORY_SCOPE_AGENT);
    __hip_atomic_fetch_add(ap + 1, v.y * w, __ATOMIC_RELAXED, __HIP_MEMORY_SCOPE_AGENT);
    __hip_atomic_fetch_add(ap + 2, v.z * w, __ATOMIC_RELAXED, __HIP_MEMORY_SCOPE_AGENT);
    __hip_atomic_fetch_add(ap + 3, v.w * w, __ATOMIC_RELAXED, __HIP_MEMORY_SCOPE_AGENT);
}

// ---------------- global mean pool ----------------

__global__ __launch_bounds__(256) void k_zero(float* __restrict__ p, int n) {
    int i = blockIdx.x * 256 + threadIdx.x;
    if (i < n) p[i] = 0.0f;
}

__global__ __launch_bounds__(256) void k_pool_accum(float* __restrict__ sums,
                                                    float* __restrict__ cnt,
                                                    const float* __restrict__ h,
                                                    const int* __restrict__ batch, int n) {
    int wid  = (blockIdx.x * 256 + threadIdx.x) >> 5;
    int lane = threadIdx.x & 31;
    if (wid >= n) return;
    int g = batch[wid];
    float4 v = *((const float4*)(h + (size_t)wid * FEAT) + lane);
    float* sp = sums + (size_t)g * FEAT + lane * 4;
    __hip_atomic_fetch_add(sp + 0, v.x, __ATOMIC_RELAXED, __HIP_MEMORY_SCOPE_AGENT);
    __hip_atomic_fetch_add(sp + 1, v.y, __ATOMIC_RELAXED, __HIP_MEMORY_SCOPE_AGENT);
    __hip_atomic_fetch_add(sp + 2, v.z, __ATOMIC_RELAXED, __HIP_MEMORY_SCOPE_AGENT);
    __hip_atomic_fetch_add(sp + 3, v.w, __ATOMIC_RELAXED, __HIP_MEMORY_SCOPE_AGENT);
    if (lane == 0)
        __hip_atomic_fetch_add(cnt + g, 1.0f, __ATOMIC_RELAXED, __HIP_MEMORY_SCOPE_AGENT);
}

__global__ __launch_bounds__(256) void k_pool_final(float* __restrict__ out,
                                                    const float* __restrict__ sums,
                                                    const float* __restrict__ cnt, int n) {
    int i = blockIdx.x * 256 + threadIdx.x;
    if (i < n) out[i] = sums[i] / fmaxf(cnt[i >> 7], 1.0f);
}

// ---------------- orchestration ----------------

extern "C" void kernel_launch(void* const* d_in, const int* in_sizes, int n_in,
                              void* d_out, int out_size, void* d_ws, size_t ws_size,
                              hipStream_t stream) {
    const float* x     = (const float*)d_in[0];
    const int*   ei    = (const int*)d_in[1];
    const int*   batch = (const int*)d_in[2];
    const float* W[3]  = {(const float*)d_in[3],  (const float*)d_in[7],  (const float*)d_in[11]};
    const float* CB[3] = {(const float*)d_in[4],  (const float*)d_in[8],  (const float*)d_in[12]};
    const float* LW[3] = {(const float*)d_in[5],  (const float*)d_in[9],  (const float*)d_in[13]};
    const float* LB[3] = {(const float*)d_in[6],  (const float*)d_in[10], (const float*)d_in[14]};

    const int nNodes  = in_sizes[0] / FEAT;     // 50000
    const int nEdges  = in_sizes[1] / 2;        // 600000
    const int poolN   = out_size;               // 128*128
    const int* srcIdx = ei;
    const int* dstIdx = ei + nEdges;

    const size_t P = (size_t)nNodes * FEAT;
    float* bufA = (float*)d_ws;                 // h / layer output
    float* bufB = bufA + P;                     // agg
    float* dinv = bufB + P;                     // deg -> dinv (nNodes)
    float* sums = dinv + nNodes;                // poolN
    float* cnt  = sums + poolN;                 // nGraphs

    const int mTiles   = (nNodes + 15) / 16;
    const int nodeBlk  = (nNodes + 255) / 256;
    const int edgeBlk  = (nEdges + 255) / 256;
    const int elemBlk  = (int)((P + 255) / 256);
    const int gemmBlk  = mTiles < 1024 ? mTiles : 1024;   // persistent grid-stride
    const int warpEBlk = (nEdges + 7) / 8;      // wave per edge
    const int warpNBlk = (nNodes + 7) / 8;      // wave per node

    // GCN normalization: deg (with self-loops) -> dinv
    k_fill_deg<<<nodeBlk, 256, 0, stream>>>(dinv, nNodes);
    k_deg_edges<<<edgeBlk, 256, 0, stream>>>(dinv, dstIdx, nEdges);
    k_deg_to_dinv<<<nodeBlk, 256, 0, stream>>>(dinv, nNodes);

    const float* layerIn = x;
    for (int l = 0; l < 3; ++l) {
        // h = layerIn @ w
        k_gemm_wmma<false, false, false><<<gemmBlk, 256, 0, stream>>>(
            layerIn, W[l], nullptr, bufA, nNodes);
        // agg = cb + h*dinv^2  (self-loops), then scatter edges
        k_init_agg<<<elemBlk, 256, 0, stream>>>(bufB, bufA, dinv, CB[l], nNodes);
        k_scatter<<<warpEBlk, 256, 0, stream>>>(bufB, bufA, dinv, srcIdx, dstIdx, nEdges);
        // out = relu(agg @ lw^T + lb)  -> bufA
        k_gemm_wmma<true, true, true><<<gemmBlk, 256, 0, stream>>>(
            bufB, LW[l], LB[l], bufA, nNodes);
        layerIn = bufA;
    }

    // global mean pool
    k_zero<<<(poolN + 128 + 255) / 256, 256, 0, stream>>>(sums, poolN + poolN / FEAT);
    k_pool_accum<<<warpNBlk, 256, 0, stream>>>(sums, cnt, bufA, batch, nNodes);
    k_pool_final<<<(poolN + 255) / 256, 256, 0, stream>>>((float*)d_out, sums, cnt, poolN);
}